// PreModel_18339510354236
// MI455X (gfx1250) — compile-verified
//
#include <hip/hip_runtime.h>
#include <hip/hip_bf16.h>

typedef float v2f __attribute__((ext_vector_type(2)));
typedef float v8f __attribute__((ext_vector_type(8)));

#define DFEAT 128

// ---------------------------------------------------------------- utilities
__global__ void zero_f(float* __restrict__ p, long long n) {
  long long i = (long long)blockIdx.x * blockDim.x + threadIdx.x;
  if (i < n) p[i] = 0.0f;
}

__global__ void zero_i(int* __restrict__ p, int n) {
  int i = blockIdx.x * blockDim.x + threadIdx.x;
  if (i < n) p[i] = 0;
}

// ---------------------------------------------------------------- degrees
__global__ void degree_count(const int* __restrict__ src, const int* __restrict__ dst,
                             float* __restrict__ degO, float* __restrict__ degI, int E) {
  int e = blockIdx.x * blockDim.x + threadIdx.x;
  if (e < E) {
    atomicAdd(&degO[src[e]], 1.0f);
    atomicAdd(&degI[dst[e]], 1.0f);
  }
}

__global__ void finalize_deg(float* __restrict__ degO, float* __restrict__ degI, int n) {
  int i = blockIdx.x * blockDim.x + threadIdx.x;
  if (i < n) {
    degO[i] = 1.0f / sqrtf(fmaxf(degO[i], 1.0f));
    degI[i] = 1.0f / sqrtf(fmaxf(degI[i], 1.0f));
  }
}

__global__ void set_mask(const int* __restrict__ mask_nodes, int* __restrict__ flag, int M) {
  int i = blockIdx.x * blockDim.x + threadIdx.x;
  if (i < M) flag[mask_nodes[i]] = 1;
}

// ---------------------------------------------------------------- row scaling
// mode 0: out = in * s[row]
// mode 1: masked rows replaced by token row, then * s[row]   (encoder input masking)
// mode 2: masked rows zeroed, then * s[row]                  (re-mask before decoder)
__global__ void scale_rows(const float* __restrict__ in, const float* __restrict__ srow,
                           const int* __restrict__ flag, const float* __restrict__ token,
                           float* __restrict__ out, int nrows, int mode) {
  int idx = blockIdx.x * blockDim.x + threadIdx.x;
  if (idx >= nrows * DFEAT) return;
  int row = idx >> 7;
  int c = idx & (DFEAT - 1);
  float v = in[idx];
  if (mode == 1 && flag[row]) v = token[c];
  if (mode == 2 && flag[row]) v = 0.0f;
  out[idx] = v * srow[row];
}

__global__ void row_scale_inplace(float* __restrict__ h, const float* __restrict__ s, int nrows) {
  int idx = blockIdx.x * blockDim.x + threadIdx.x;
  if (idx >= nrows * DFEAT) return;
  h[idx] *= s[idx >> 7];
}

// ---------------------------------------------------------------- SpMM (scatter-add)
// One wave32 per edge. float4 gather (512B coalesced per wave) + 4 f32 atomics/lane.
// Node features (25.6 MB) are L2-resident (192 MB), so atomics resolve in L2.
__global__ void spmm_atomic(const float* __restrict__ feat, const int* __restrict__ src,
                            const int* __restrict__ dst, float* __restrict__ accum, int E) {
  int wave = (int)(((unsigned)blockIdx.x * blockDim.x + threadIdx.x) >> 5);
  int lane = threadIdx.x & 31;
  if (wave >= E) return;
  int s = src[wave];
  int d = dst[wave];
  const float4 v = ((const float4*)(feat + (size_t)s * DFEAT))[lane];
  float* od = accum + (size_t)d * DFEAT + lane * 4;
  atomicAdd(od + 0, v.x);
  atomicAdd(od + 1, v.y);
  atomicAdd(od + 2, v.z);
  atomicAdd(od + 3, v.w);
}

// ---------------------------------------------------------------- WMMA f32 GEMM
// Out[m][j] = sum_k In[m][k] * W[j][k]  (+ bias[j]), all dims 128 in K/J.
// Block = 256 threads = 8 waves; block owns one 16-row tile of In (staged in LDS,
// contiguous + coalesced), wave w computes j-tile w with V_WMMA_F32_16X16X4_F32.
__global__ __launch_bounds__(256) void gemm_wmma_f32(
    const float* __restrict__ In, const float* __restrict__ W,
    const float* __restrict__ bias, float* __restrict__ Out, int nrows) {
  __shared__ float Atile[16][DFEAT + 4];  // +4 pad: A-operand reads hit distinct banks
  const int m0 = blockIdx.x * 16;
  const int tid = threadIdx.x;

  // 16 rows x 128 cols are contiguous in memory -> perfectly coalesced fill
  for (int i = tid; i < 16 * DFEAT; i += 256) {
    int r = i >> 7, c = i & (DFEAT - 1);
    int m = m0 + r;
    Atile[r][c] = (m < nrows) ? In[(size_t)m * DFEAT + c] : 0.0f;
  }
  __syncthreads();

  const int wave = tid >> 5;   // j-tile index 0..7
  const int lane = tid & 31;
  const int hi = lane >> 4;    // selects K pair {0,1} vs {2,3}
  const int lo = lane & 15;    // A: row M; B: col N
  const int j0 = wave * 16;
  const float* wrow = W + (size_t)(j0 + lo) * DFEAT;

  v8f acc = {};
  #pragma unroll
  for (int k = 0; k < DFEAT; k += 4) {
    v2f a, b;
    const int ka = k + 2 * hi;
    a.x = Atile[lo][ka];
    a.y = Atile[lo][ka + 1];
    b.x = wrow[ka];
    b.y = wrow[ka + 1];
    acc = __builtin_amdgcn_wmma_f32_16x16x4_f32(
        /*neg_a=*/false, a, /*neg_b=*/false, b,
        /*c_mod=*/(short)0, acc, /*reuse_a=*/false, /*reuse_b=*/false);
  }

  #pragma unroll
  for (int r = 0; r < 8; ++r) {
    int m = m0 + r + 8 * hi;
    if (m < nrows) {
      float v = acc[r];
      if (bias) v += bias[j0 + lo];
      Out[(size_t)m * DFEAT + j0 + lo] = v;
    }
  }
}

// ---------------------------------------------------------------- LN + PReLU (wave per row)
__global__ void post_ln_prelu(float* __restrict__ h, const float* __restrict__ degI_rs,
                              const float* __restrict__ g, const float* __restrict__ be,
                              const float* __restrict__ a, int nrows) {
  int row = (int)(((unsigned)blockIdx.x * blockDim.x + threadIdx.x) >> 5);
  int lane = threadIdx.x & 31;
  if (row >= nrows) return;
  float* p = h + (size_t)row * DFEAT;
  const float s = degI_rs[row];
  float v[4];
  float sum = 0.0f;
  #pragma unroll
  for (int i = 0; i < 4; ++i) { v[i] = p[lane + 32 * i] * s; sum += v[i]; }
  for (int off = 16; off > 0; off >>= 1) sum += __shfl_xor(sum, off, 32);
  const float mu = sum * (1.0f / DFEAT);
  float sq = 0.0f;
  #pragma unroll
  for (int i = 0; i < 4; ++i) { float d = v[i] - mu; sq += d * d; }
  for (int off = 16; off > 0; off >>= 1) sq += __shfl_xor(sq, off, 32);
  const float inv = 1.0f / sqrtf(sq * (1.0f / DFEAT) + 1e-5f);
  const float alpha = a[0];
  #pragma unroll
  for (int i = 0; i < 4; ++i) {
    int c = lane + 32 * i;
    float y = (v[i] - mu) * inv * g[c] + be[c];
    p[c] = (y >= 0.0f) ? y : alpha * y;
  }
}

// ---------------------------------------------------------------- SCE loss (wave per masked row)
__global__ void sce_loss(const float* __restrict__ recon, const float* __restrict__ x,
                         const int* __restrict__ mask_nodes, int M, float* __restrict__ out) {
  int i = (int)(((unsigned)blockIdx.x * blockDim.x + threadIdx.x) >> 5);
  int lane = threadIdx.x & 31;
  if (i >= M) return;
  int n = mask_nodes[i];
  const float* r = recon + (size_t)n * DFEAT;
  const float* xr = x + (size_t)n * DFEAT;
  float rr = 0.0f, xx = 0.0f, rx = 0.0f;
  for (int c = lane; c < DFEAT; c += 32) {
    float av = r[c], bv = xr[c];
    rr += av * av; xx += bv * bv; rx += av * bv;
  }
  for (int off = 16; off > 0; off >>= 1) {
    rr += __shfl_xor(rr, off, 32);
    xx += __shfl_xor(xx, off, 32);
    rx += __shfl_xor(rx, off, 32);
  }
  if (lane == 0) {
    float nr = fmaxf(sqrtf(rr), 1e-12f);
    float nx = fmaxf(sqrtf(xx), 1e-12f);
    float t = 1.0f - rx / (nr * nx);
    atomicAdd(out, t * t / (float)M);
  }
}

// ---------------------------------------------------------------- driver
extern "C" void kernel_launch(void* const* d_in, const int* in_sizes, int n_in,
                              void* d_out, int out_size, void* d_ws, size_t ws_size,
                              hipStream_t stream) {
  const float* x     = (const float*)d_in[0];
  const float* token = (const float*)d_in[1];
  const float* W1 = (const float*)d_in[2];
  const float* b1 = (const float*)d_in[3];
  const float* g1 = (const float*)d_in[4];
  const float* be1 = (const float*)d_in[5];
  const float* a1 = (const float*)d_in[6];
  const float* W2 = (const float*)d_in[7];
  const float* b2 = (const float*)d_in[8];
  const float* g2 = (const float*)d_in[9];
  const float* be2 = (const float*)d_in[10];
  const float* a2 = (const float*)d_in[11];
  const float* We2d = (const float*)d_in[12];
  const float* Wd = (const float*)d_in[13];
  const float* bd = (const float*)d_in[14];
  const int* src = (const int*)d_in[15];
  const int* dst = (const int*)d_in[16];
  const int* mask_nodes = (const int*)d_in[17];

  const int N = in_sizes[0] / DFEAT;
  const int E = in_sizes[15];
  const int M = in_sizes[17];

  // workspace layout (~78 MB): degO | degI | flag | bufA | bufB | bufC
  float* degO = (float*)d_ws;
  float* degI = degO + N;
  int* flag = (int*)(degI + N);
  float* bufA = (float*)(flag + N);
  float* bufB = bufA + (size_t)N * DFEAT;
  float* bufC = bufB + (size_t)N * DFEAT;

  const int TB = 256;
  const long long ND = (long long)N * DFEAT;
  dim3 blk(TB);
  dim3 gND((unsigned)((ND + TB - 1) / TB));
  dim3 gN((unsigned)((N + TB - 1) / TB));
  dim3 gE((unsigned)((E + TB - 1) / TB));
  dim3 gEw((unsigned)(((long long)E * 32 + TB - 1) / TB));  // wave per edge
  dim3 gNw((unsigned)(((long long)N * 32 + TB - 1) / TB));  // wave per row
  dim3 gMw((unsigned)(((long long)M * 32 + TB - 1) / TB));  // wave per masked row
  dim3 gG((unsigned)((N + 15) / 16));                       // 16-row tile per block

  // degrees + mask flags
  zero_f<<<dim3((unsigned)((2LL * N + TB - 1) / TB)), blk, 0, stream>>>(degO, 2LL * N);
  zero_i<<<gN, blk, 0, stream>>>(flag, N);
  degree_count<<<gE, blk, 0, stream>>>(src, dst, degO, degI, E);
  finalize_deg<<<gN, blk, 0, stream>>>(degO, degI, N);
  set_mask<<<dim3((unsigned)((M + TB - 1) / TB)), blk, 0, stream>>>(mask_nodes, flag, M);

  // encoder layer 1: mask+scale -> spmm -> gemm(W1)+b1 -> degI scale + LN + PReLU
  scale_rows<<<gND, blk, 0, stream>>>(x, degO, flag, token, bufA, N, 1);
  zero_f<<<gND, blk, 0, stream>>>(bufB, ND);
  spmm_atomic<<<gEw, blk, 0, stream>>>(bufA, src, dst, bufB, E);
  gemm_wmma_f32<<<gG, blk, 0, stream>>>(bufB, W1, b1, bufC, N);
  post_ln_prelu<<<gNw, blk, 0, stream>>>(bufC, degI, g1, be1, a1, N);

  // encoder layer 2
  scale_rows<<<gND, blk, 0, stream>>>(bufC, degO, flag, token, bufA, N, 0);
  zero_f<<<gND, blk, 0, stream>>>(bufB, ND);
  spmm_atomic<<<gEw, blk, 0, stream>>>(bufA, src, dst, bufB, E);
  gemm_wmma_f32<<<gG, blk, 0, stream>>>(bufB, W2, b2, bufA, N);
  post_ln_prelu<<<gNw, blk, 0, stream>>>(bufA, degI, g2, be2, a2, N);

  // encoder_to_decoder (no bias) -> re-mask -> decoder GraphConv
  gemm_wmma_f32<<<gG, blk, 0, stream>>>(bufA, We2d, nullptr, bufC, N);
  scale_rows<<<gND, blk, 0, stream>>>(bufC, degO, flag, token, bufB, N, 2);
  zero_f<<<gND, blk, 0, stream>>>(bufA, ND);
  spmm_atomic<<<gEw, blk, 0, stream>>>(bufB, src, dst, bufA, E);
  gemm_wmma_f32<<<gG, blk, 0, stream>>>(bufA, Wd, bd, bufB, N);
  row_scale_inplace<<<gND, blk, 0, stream>>>(bufB, degI, N);

  // SCE loss over masked rows
  zero_f<<<dim3(1), dim3(1), 0, stream>>>((float*)d_out, 1);
  sce_loss<<<gMw, blk, 0, stream>>>(bufB, x, mask_nodes, M, (float*)d_out);
}